// TreeCrossEntropyLoss_18305150616186
// MI455X (gfx1250) — compile-verified
//
#include <hip/hip_runtime.h>

// Problem constants (fixed by the reference)
#define Bn 8
#define Cn 20
#define Hn 512
#define Wn 512
#define HW (Hn * Wn)            // 262144
#define NPIX (Bn * HW)          // 2097152
#define NBLOCKS 2048
#define THREADS 256
#define PIX_PER_BLOCK (NPIX / NBLOCKS)            // 1024
#define TILES_PER_WAVE (PIX_PER_BLOCK / (8 * 16)) // 8

#define CLAMP_LO 1e-7f
#define CLAMP_HI 0.9999999f

typedef __attribute__((ext_vector_type(2))) float v2f;
typedef __attribute__((ext_vector_type(8))) float v8f;

// Membership matrix rows (the WMMA A matrix): row m, channel k.
// m=0: all 20 channels; m=1,2: the two groups of 10; m=3..6: groups of 5; m>=7: zero.
__device__ __forceinline__ float memb(int m, int k) {
  bool v;
  if      (m == 0) v = (k < 20);
  else if (m == 1) v = (k < 10);
  else if (m == 2) v = (k >= 10) && (k < 20);
  else if (m == 3) v = (k < 5);
  else if (m == 4) v = (k >= 5)  && (k < 10);
  else if (m == 5) v = (k >= 10) && (k < 15);
  else if (m == 6) v = (k >= 15) && (k < 20);
  else             v = false;
  return v ? 1.0f : 0.0f;
}

__global__ __launch_bounds__(THREADS) void tree_ce_partial(
    const float* __restrict__ logits, const int* __restrict__ targets,
    float* __restrict__ ws_partials) {
  const int tid  = threadIdx.x;
  const int lane = tid & 31;
  const int wv   = tid >> 5;
  const int n    = lane & 15;  // pixel slot within the 16-pixel tile (also A row M)
  const int kh   = lane >> 4;  // K-half: this half-wave owns channels with bit1 == kh

  // A-matrix fragments (constant): per ISA layout for 32-bit A 16x4,
  // lane L holds row M=(L&15); VGPR0 = K=(4s+2*kh+0), VGPR1 = K=(4s+2*kh+1).
  float aK0[5], aK1[5];
  #pragma unroll
  for (int s = 0; s < 5; ++s) {
    aK0[s] = memb(n, 4 * s + 2 * kh + 0);
    aK1[s] = memb(n, 4 * s + 2 * kh + 1);
  }

  float sum1 = 0.0f, sum2 = 0.0f, sum3 = 0.0f;

  #pragma unroll 1
  for (int it = 0; it < TILES_PER_WAVE; ++it) {
    const int P0 = blockIdx.x * PIX_PER_BLOCK + it * (8 * 16) + wv * 16;
    const int bi = P0 / HW;              // tile never crosses a batch boundary
    const int q  = (P0 - bi * HW) + n;   // in-image pixel for this lane
    const float* base = logits + (size_t)bi * (Cn * HW) + q;

    // Both half-waves read the same target (one 64B segment, broadcast).
    const int t = __builtin_nontemporal_load(targets + P0 + n);

    // Streaming, non-temporal loads of this lane's 10 channels (lane-contiguous).
    float x[10];
    #pragma unroll
    for (int s = 0; s < 5; ++s) {
      #pragma unroll
      for (int j = 0; j < 2; ++j) {
        const int c = 4 * s + 2 * kh + j;
        x[2 * s + j] = __builtin_nontemporal_load(base + (size_t)c * HW);
      }
    }

    // Per-pixel max across both half-waves (numerically-safe softmax shift).
    float mymax = x[0];
    #pragma unroll
    for (int i = 1; i < 10; ++i) mymax = fmaxf(mymax, x[i]);
    const float m = fmaxf(mymax, __shfl_xor(mymax, 16, 32));

    // exp values; capture e_t via selects (no divergence, no scratch indexing).
    float e[10];
    float et_c = 0.0f;
    #pragma unroll
    for (int s = 0; s < 5; ++s) {
      #pragma unroll
      for (int j = 0; j < 2; ++j) {
        const int c  = 4 * s + 2 * kh + j;
        const float ev = __expf(x[2 * s + j] - m);
        e[2 * s + j] = ev;
        et_c = (c == t) ? ev : et_c;
      }
    }
    const float et = et_c + __shfl_xor(et_c, 16, 32);

    // D[16 sums x 16 pixels] = A(membership) x B(exp), chained K=4 steps, K=20.
    // B layout (32-bit 4x16): lanes 0-15 rows K=0,1; lanes 16-31 rows K=2,3 —
    // exactly the channels this lane loaded.
    v8f acc = {};
    #pragma unroll
    for (int s = 0; s < 5; ++s) {
      v2f a; a.x = aK0[s];     a.y = aK1[s];
      v2f b; b.x = e[2 * s];   b.y = e[2 * s + 1];
      acc = __builtin_amdgcn_wmma_f32_16x16x4_f32(
          /*neg_a=*/false, a, /*neg_b=*/false, b,
          /*c_mod=*/(short)0, acc, /*reuse_a=*/false, /*reuse_b=*/false);
    }

    // Lane n (n<16) now holds D[M=0..7][N=n] = all branch sums for its pixel.
    const float S20 = acc[0];
    const float S10 = (t < 10) ? acc[1] : acc[2];
    const float S5  = (t < 5) ? acc[3]
                    : ((t < 10) ? acc[4] : ((t < 15) ? acc[5] : acc[6]));
    const float inv = 1.0f / S20;

    const float l1 = __logf(fminf(fmaxf(S10 * inv, CLAMP_LO), CLAMP_HI));
    const float l2 = __logf(fminf(fmaxf(S5  * inv, CLAMP_LO), CLAMP_HI));
    const float l3 = __logf(fminf(fmaxf(et  * inv, CLAMP_LO), CLAMP_HI));

    const bool act = (lane < 16);  // lanes 16-31 hold zero D rows; selects avoid NaN
    sum1 += act ? l1 : 0.0f;
    sum2 += act ? l2 : 0.0f;
    sum3 += act ? l3 : 0.0f;
  }

  // Deterministic fixed-order block reduction.
  __shared__ float sh[3][THREADS];
  sh[0][tid] = sum1; sh[1][tid] = sum2; sh[2][tid] = sum3;
  __syncthreads();
  for (int off = THREADS / 2; off > 0; off >>= 1) {
    if (tid < off) {
      sh[0][tid] += sh[0][tid + off];
      sh[1][tid] += sh[1][tid + off];
      sh[2][tid] += sh[2][tid + off];
    }
    __syncthreads();
  }
  if (tid == 0) {
    ws_partials[0 * NBLOCKS + blockIdx.x] = sh[0][0];
    ws_partials[1 * NBLOCKS + blockIdx.x] = sh[1][0];
    ws_partials[2 * NBLOCKS + blockIdx.x] = sh[2][0];
  }
}

__global__ __launch_bounds__(THREADS) void tree_ce_finalize(
    const float* __restrict__ ws_partials, float* __restrict__ out) {
  __shared__ float sh[3][THREADS];
  const int tid = threadIdx.x;
  #pragma unroll
  for (int l = 0; l < 3; ++l) {
    float s = 0.0f;
    for (int i = tid; i < NBLOCKS; i += THREADS) s += ws_partials[l * NBLOCKS + i];
    sh[l][tid] = s;
  }
  __syncthreads();
  for (int off = THREADS / 2; off > 0; off >>= 1) {
    if (tid < off) {
      sh[0][tid] += sh[0][tid + off];
      sh[1][tid] += sh[1][tid + off];
      sh[2][tid] += sh[2][tid + off];
    }
    __syncthreads();
  }
  if (tid == 0) {
    const float inv_n = 1.0f / (float)NPIX;
    const float L1 = -sh[0][0] * inv_n;
    const float L2 = -sh[1][0] * inv_n;
    const float L3 = -sh[2][0] * inv_n;
    out[0] = L1 + L2 + L3;  // total loss
    out[1] = L1;            // level losses
    out[2] = L2;
    out[3] = L3;
  }
}

extern "C" void kernel_launch(void* const* d_in, const int* in_sizes, int n_in,
                              void* d_out, int out_size, void* d_ws, size_t ws_size,
                              hipStream_t stream) {
  const float* logits  = (const float*)d_in[0];
  const int*   targets = (const int*)d_in[1];
  float* ws  = (float*)d_ws;   // needs 3 * NBLOCKS * 4 = 24 KB
  float* out = (float*)d_out;  // 4 floats: [loss, l1, l2, l3]

  tree_ce_partial<<<NBLOCKS, THREADS, 0, stream>>>(logits, targets, ws);
  tree_ce_finalize<<<1, THREADS, 0, stream>>>(ws, out);
}